// ResidualActivationFunction_83330955477908
// MI455X (gfx1250) — compile-verified
//
#include <hip/hip_runtime.h>

// Native 16-byte vector type so __builtin_nontemporal_load/store accept it
typedef float vf4 __attribute__((ext_vector_type(4)));

__device__ __forceinline__ float eval_one(float xf,
    float c0, float c1, float c2, float c3, float c4,
    float c5, float c6, float c7, float c8, float c9,
    float wb, float ws)
{
    // ---- interval index + local coordinate (uniform knots t[m]=m/13) ----
    float t13 = xf * 13.0f;
    float fi  = floorf(t13);
    fi = fminf(fmaxf(fi, 3.0f), 9.0f);   // clip for extrapolation, as reference
    float u = t13 - fi;                  // may be outside [0,1) at the ends
    int   s = (int)fi - 3;               // window start in [0,6]

    // ---- select c[s..s+3] with a register barrel shift (15 cndmask) ----
    bool b4 = (s & 4) != 0;              // if set, remaining shift <= 2
    float e0 = b4 ? c4 : c0;
    float e1 = b4 ? c5 : c1;
    float e2 = b4 ? c6 : c2;
    float e3 = b4 ? c7 : c3;
    float e4 = b4 ? c8 : c4;
    float e5 = b4 ? c9 : c5;
    float e6 = c6;                       // only consumed when !b4
    bool b2 = (s & 2) != 0;
    float f0 = b2 ? e2 : e0;
    float f1 = b2 ? e3 : e1;
    float f2 = b2 ? e4 : e2;
    float f3 = b2 ? e5 : e3;
    float f4 = b2 ? e6 : e4;
    bool b1 = (s & 1) != 0;
    float d0 = b1 ? f1 : f0;
    float d1 = b1 ? f2 : f1;
    float d2 = b1 ? f3 : f2;
    float d3 = b1 ? f4 : f3;

    // ---- De Boor, degree 3, uniform spacing: alpha=(u+3-j)/(4-r) ----
    const float I3 = 0.33333333333333333f;
    float a = u * I3;                              // r=1
    d3 = fmaf(a,                 d3 - d2, d2);
    d2 = fmaf(a + I3,            d2 - d1, d1);
    d1 = fmaf(a + 0.66666666666666667f, d1 - d0, d0);
    float h = u * 0.5f;                            // r=2
    d3 = fmaf(h,        d3 - d2, d2);
    d2 = fmaf(h + 0.5f, d2 - d1, d1);
    d3 = fmaf(u,        d3 - d2, d2);              // r=3 -> spline value

    // ---- silu(x) = x / (1 + exp(-x)) ----
    float ex    = __expf(-xf);                     // v_exp_f32 (trans pipe)
    float sig   = __builtin_amdgcn_rcpf(1.0f + ex);// v_rcp_f32
    float basis = xf * sig;

    return fmaf(wb, basis, ws * d3);
}

#define TPB   256
#define VPT   4            // float4 tiles per thread
#define TILE  (TPB * VPT)  // float4 tiles per block

__global__ __launch_bounds__(TPB)
void residual_spline_kernel(const float* __restrict__ x,
                            const float* __restrict__ c,
                            const float* __restrict__ wbp,
                            const float* __restrict__ wsp,
                            float* __restrict__ out,
                            int nv4, int n)
{
    // Uniform scalar loads (compiler lowers to s_load_b32): 10 coefs + weights
    const float c0 = c[0], c1 = c[1], c2 = c[2], c3 = c[3], c4 = c[4];
    const float c5 = c[5], c6 = c[6], c7 = c[7], c8 = c[8], c9 = c[9];
    const float wb = wbp[0], ws = wsp[0];

    const vf4* __restrict__ xv = (const vf4*)x;
    vf4* __restrict__ ov = (vf4*)out;

    int base = blockIdx.x * TILE + threadIdx.x;

    if (base + (VPT - 1) * TPB < nv4) {
        // Fast path: full tile. Issue all NT b128 loads first, then compute.
        vf4 v[VPT];
#pragma unroll
        for (int j = 0; j < VPT; ++j)
            v[j] = __builtin_nontemporal_load(xv + base + j * TPB);
#pragma unroll
        for (int j = 0; j < VPT; ++j) {
            vf4 r;
            r.x = eval_one(v[j].x, c0,c1,c2,c3,c4,c5,c6,c7,c8,c9, wb, ws);
            r.y = eval_one(v[j].y, c0,c1,c2,c3,c4,c5,c6,c7,c8,c9, wb, ws);
            r.z = eval_one(v[j].z, c0,c1,c2,c3,c4,c5,c6,c7,c8,c9, wb, ws);
            r.w = eval_one(v[j].w, c0,c1,c2,c3,c4,c5,c6,c7,c8,c9, wb, ws);
            __builtin_nontemporal_store(r, ov + base + j * TPB);
        }
    } else {
        // Guarded edge tile
#pragma unroll
        for (int j = 0; j < VPT; ++j) {
            int idx = base + j * TPB;
            if (idx < nv4) {
                vf4 v = __builtin_nontemporal_load(xv + idx);
                vf4 r;
                r.x = eval_one(v.x, c0,c1,c2,c3,c4,c5,c6,c7,c8,c9, wb, ws);
                r.y = eval_one(v.y, c0,c1,c2,c3,c4,c5,c6,c7,c8,c9, wb, ws);
                r.z = eval_one(v.z, c0,c1,c2,c3,c4,c5,c6,c7,c8,c9, wb, ws);
                r.w = eval_one(v.w, c0,c1,c2,c3,c4,c5,c6,c7,c8,c9, wb, ws);
                __builtin_nontemporal_store(r, ov + idx);
            }
        }
        // Scalar tail (n not divisible by 4) handled by block 0
        if (blockIdx.x == 0) {
            int rem = n - (nv4 << 2);
            if ((int)threadIdx.x < rem) {
                int idx = (nv4 << 2) + threadIdx.x;
                out[idx] = eval_one(x[idx],
                                    c0,c1,c2,c3,c4,c5,c6,c7,c8,c9, wb, ws);
            }
        }
    }
}

extern "C" void kernel_launch(void* const* d_in, const int* in_sizes, int n_in,
                              void* d_out, int out_size, void* d_ws, size_t ws_size,
                              hipStream_t stream)
{
    const float* x  = (const float*)d_in[0];   // (N,) fp32 in [0,1)
    const float* c  = (const float*)d_in[1];   // (10,) fp32
    const float* wb = (const float*)d_in[2];   // (1,) fp32
    const float* ws = (const float*)d_in[3];   // (1,) fp32
    // d_in[4] = degree (always 3; baked into the kernel)
    float* out = (float*)d_out;

    int n   = in_sizes[0];
    int nv4 = n >> 2;
    int blocks = (nv4 + TILE - 1) / TILE;
    if (blocks < 1) blocks = 1;

    residual_spline_kernel<<<dim3(blocks), dim3(TPB), 0, stream>>>(
        x, c, wb, ws, out, nv4, n);
}